// ConceptFusionModule_88330297410158
// MI455X (gfx1250) — compile-verified
//
#include <hip/hip_runtime.h>
#include <math.h>

typedef __attribute__((ext_vector_type(2))) float v2f;
typedef __attribute__((ext_vector_type(8))) float v8f;

constexpr int Bb = 2, Nn = 4096, Ll = 256, Dd = 2048, Cc = 6;

// ---------------------------------------------------------------- utilities
__global__ void zero_kernel(float* __restrict__ p, int n) {
    int i = blockIdx.x * blockDim.x + threadIdx.x;
    if (i < n) p[i] = 0.0f;
}

// ------------------------------------------------- 1) cluster softmax weights
// one wave32 per (b,l) row: logits[c] = dot(text_emb[b,l,:], Wc[c,:]), softmax over C
__global__ void cluster_weights_kernel(const float* __restrict__ te,
                                       const float* __restrict__ Wc,
                                       float* __restrict__ w) {
    int wave = (blockIdx.x * blockDim.x + threadIdx.x) >> 5;   // 0 .. B*L-1
    int lane = threadIdx.x & 31;
    if (wave >= Bb * Ll) return;
    const float* row = te + (size_t)wave * Dd;
    float acc[Cc];
#pragma unroll
    for (int c = 0; c < Cc; ++c) acc[c] = 0.0f;
    for (int d = lane; d < Dd; d += 32) {
        float t = row[d];
#pragma unroll
        for (int c = 0; c < Cc; ++c) acc[c] = fmaf(t, Wc[c * Dd + d], acc[c]);
    }
#pragma unroll
    for (int c = 0; c < Cc; ++c)
#pragma unroll
        for (int off = 16; off > 0; off >>= 1)
            acc[c] += __shfl_xor(acc[c], off, 32);
    if (lane == 0) {
        float mx = acc[0];
#pragma unroll
        for (int c = 1; c < Cc; ++c) mx = fmaxf(mx, acc[c]);
        float e[Cc], s = 0.0f;
#pragma unroll
        for (int c = 0; c < Cc; ++c) { e[c] = expf(acc[c] - mx); s += e[c]; }
        float inv = 1.0f / s;
#pragma unroll
        for (int c = 0; c < Cc; ++c) w[wave * Cc + c] = e[c] * inv;
    }
}

// ------------------------------------------------- 2) concept pooling -> A1 rows 0..11
// A1 is a 16 x D zero-padded A-matrix for the WMMA GEMM; row m = b*C + c
__global__ void concept_reps_kernel(const float* __restrict__ te,
                                    const float* __restrict__ w,
                                    float* __restrict__ A1) {
    int t = blockIdx.x * blockDim.x + threadIdx.x;   // 0 .. B*D-1
    int b = t >> 11;          // D = 2048
    int d = t & (Dd - 1);
    float acc[Cc];
#pragma unroll
    for (int c = 0; c < Cc; ++c) acc[c] = 0.0f;
    const float* teb = te + (size_t)b * Ll * Dd + d;
    const float* wb  = w + b * Ll * Cc;
    for (int l = 0; l < Ll; ++l) {
        float tv = teb[(size_t)l * Dd];
#pragma unroll
        for (int c = 0; c < Cc; ++c) acc[c] = fmaf(wb[l * Cc + c], tv, acc[c]);
    }
#pragma unroll
    for (int c = 0; c < Cc; ++c) A1[(size_t)(b * Cc + c) * Dd + d] = acc[c];
}

// ------------------------------------------------- 3) skinny GEMM: Out[m,n] = sum_d A[m,d]*W[n,d]
// A: 16 x K (zero padded), W: Nfull x K row-major (used transposed), one wave per 16-col tile.
// f32 WMMA layouts: A(16x4) lane loads A[lane&15, k + 2*(lane>>4) + {0,1}];
// B(4x16)=W^T slice: lane loads W[n0 + (lane&15), k + 2*(lane>>4) + {0,1}] -> contiguous float2.
__global__ void wmma_gemm_abt_kernel(const float* __restrict__ A,
                                     const float* __restrict__ W,
                                     float* __restrict__ Out,
                                     int K, int m_store, int ldOut) {
    int n0   = blockIdx.x * 16;
    int lane = threadIdx.x;          // blockDim.x == 32, EXEC all ones for WMMA
    int half  = lane >> 4;
    int idx16 = lane & 15;
    const float* aptr = A + (size_t)idx16 * K + half * 2;
    const float* bptr = W + (size_t)(n0 + idx16) * K + half * 2;
    v8f acc = {0.f, 0.f, 0.f, 0.f, 0.f, 0.f, 0.f, 0.f};
#pragma unroll 8
    for (int k = 0; k < K; k += 4) {
        v2f a = *(const v2f*)(aptr + k);
        v2f b = *(const v2f*)(bptr + k);
        acc = __builtin_amdgcn_wmma_f32_16x16x4_f32(false, a, false, b,
                                                    (short)0, acc, false, false);
    }
    // D layout: VGPR j -> row (j + 8*half), col n0 + idx16
#pragma unroll
    for (int j = 0; j < 8; ++j) {
        int m = j + half * 8;
        if (m < m_store) Out[(size_t)m * ldOut + n0 + idx16] = acc[j];
    }
}

// ------------------------------------------------- 4) blend + variance gate + RMSNorm -> A2 rows 0..1
__global__ void fuse_norm_kernel(const float* __restrict__ v,
                                 const float* __restrict__ blend_w,
                                 const float* __restrict__ surreal_gate,
                                 const int*   __restrict__ is_surreal,
                                 const float* __restrict__ norm_weight,
                                 float* __restrict__ A2) {
    int b   = blockIdx.x;
    int tid = threadIdx.x;                      // 256 threads
    // blend softmax over 6 (cheap, every thread)
    float bw[Cc];
    float mx = blend_w[0];
#pragma unroll
    for (int c = 1; c < Cc; ++c) mx = fmaxf(mx, blend_w[c]);
    float s = 0.0f;
#pragma unroll
    for (int c = 0; c < Cc; ++c) { bw[c] = expf(blend_w[c] - mx); s += bw[c]; }
    float inv = 1.0f / s;
#pragma unroll
    for (int c = 0; c < Cc; ++c) bw[c] *= inv;

    float gate = 1.0f / (1.0f + expf(-surreal_gate[0]));
    int   sur  = is_surreal[0];

    constexpr int PT = Dd / 256;                // 8 elements per thread
    float fused[PT];
    float ss = 0.0f;
#pragma unroll
    for (int i = 0; i < PT; ++i) {
        int d = tid + i * 256;
        float vv[Cc], mean = 0.0f;
#pragma unroll
        for (int c = 0; c < Cc; ++c) { vv[c] = v[(size_t)(b * Cc + c) * Dd + d]; mean += vv[c]; }
        mean *= (1.0f / Cc);
        float var = 0.0f;
#pragma unroll
        for (int c = 0; c < Cc; ++c) { float t = vv[c] - mean; var = fmaf(t, t, var); }
        var *= (1.0f / (Cc - 1));               // unbiased (ddof=1)
        float f = 0.0f;
#pragma unroll
        for (int c = 0; c < Cc; ++c) f = fmaf(bw[c], vv[c], f);
        if (sur) f = fmaf(gate * 0.3f, var, f);
        fused[i] = f;
        ss = fmaf(f, f, ss);
    }
    __shared__ float red[256];
    red[tid] = ss;
    __syncthreads();
    for (int st = 128; st > 0; st >>= 1) {
        if (tid < st) red[tid] += red[tid + st];
        __syncthreads();
    }
    float rms = rsqrtf(red[0] / (float)Dd + 1e-6f);
#pragma unroll
    for (int i = 0; i < PT; ++i) {
        int d = tid + i * 256;
        A2[(size_t)b * Dd + d] = fused[i] * rms * norm_weight[d];
    }
}

// ------------------------------------------------- 5) streaming add (the only O(B*N*D) work)
__global__ void add_bcast_kernel(const float4* __restrict__ x,
                                 const float4* __restrict__ out_row,
                                 float4* __restrict__ out) {
    size_t i = (size_t)blockIdx.x * blockDim.x + threadIdx.x;   // over B*N*D/4
    int b  = (int)(i / ((size_t)Nn * Dd / 4));
    int e4 = (int)(i & (Dd / 4 - 1));
    float4 xv = x[i];
    float4 r  = out_row[b * (Dd / 4) + e4];
    float4 o;
    o.x = xv.x + r.x; o.y = xv.y + r.y; o.z = xv.z + r.z; o.w = xv.w + r.w;
    out[i] = o;
}

// ---------------------------------------------------------------- launcher
extern "C" void kernel_launch(void* const* d_in, const int* in_sizes, int n_in,
                              void* d_out, int out_size, void* d_ws, size_t ws_size,
                              hipStream_t stream) {
    const float* x     = (const float*)d_in[0];
    const float* te    = (const float*)d_in[1];
    const float* Wc    = (const float*)d_in[2];
    // d_in[3] = Wq, d_in[4] = Wk: drop out of the forward math (softmax over 1 key)
    const float* Wv    = (const float*)d_in[5];
    const float* Wo    = (const float*)d_in[6];
    const float* blend = (const float*)d_in[7];
    const float* sg    = (const float*)d_in[8];
    const float* nw    = (const float*)d_in[9];
    const int*   isr   = (const int*)d_in[10];
    float* out = (float*)d_out;

    char* ws = (char*)d_ws;
    float* w_clu   = (float*)(ws);                 // B*L*C   = 12288 B (pad to 16 KB)
    float* A1      = (float*)(ws + 16384);         // 16 x D  = 131072 B
    float* vbuf    = (float*)(ws + 147456);        // 12 x D  =  98304 B
    float* A2      = (float*)(ws + 245760);        // 16 x D  = 131072 B
    float* out_row = (float*)(ws + 376832);        //  2 x D  =  16384 B

    // zero the padded A-matrices (rows beyond the real M must be 0)
    zero_kernel<<<(16 * Dd + 255) / 256, 256, 0, stream>>>(A1, 16 * Dd);
    zero_kernel<<<(16 * Dd + 255) / 256, 256, 0, stream>>>(A2, 16 * Dd);

    // 1) soft cluster weights: 512 rows, one wave each (64 blocks x 8 waves)
    cluster_weights_kernel<<<(Bb * Ll * 32) / 256, 256, 0, stream>>>(te, Wc, w_clu);

    // 2) concept pooling into padded A1 rows 0..11
    concept_reps_kernel<<<(Bb * Dd) / 256, 256, 0, stream>>>(te, w_clu, A1);

    // 3) v = concept_reps @ Wv^T   (M=12 padded to 16, K=N=2048) — f32 WMMA
    wmma_gemm_abt_kernel<<<Dd / 16, 32, 0, stream>>>(A1, Wv, vbuf, Dd, Bb * Cc, Dd);

    // 4) blend + variance gate + RMSNorm -> A2 rows 0..1
    fuse_norm_kernel<<<Bb, 256, 0, stream>>>(vbuf, blend, sg, isr, nw, A2);

    // 5) out_row = y @ Wo^T  (M=2 padded to 16) — f32 WMMA
    wmma_gemm_abt_kernel<<<Dd / 16, 32, 0, stream>>>(A2, Wo, out_row, Dd, Bb, Dd);

    // 6) out = x + broadcast(out_row): 134 MB streamed, the bandwidth-bound part
    size_t nvec = (size_t)Bb * Nn * Dd / 4;
    add_bcast_kernel<<<(unsigned)(nvec / 256), 256, 0, stream>>>(
        (const float4*)x, (const float4*)out_row, (float4*)out);
}